// TurnEmbedding_50053548867731
// MI455X (gfx1250) — compile-verified
//
#include <hip/hip_runtime.h>
#include <hip/hip_bf16.h>

typedef __attribute__((ext_vector_type(2))) float v2f;
typedef __attribute__((ext_vector_type(4))) float v4f;
typedef __attribute__((ext_vector_type(8))) float v8f;

#define OUT_DIM 128
#define NT 8      // 128 / 16 output tiles per wave
#define KSTEPS 4  // K = 16 folded (t,p), 4 K-steps of 4 (one per turn t)

// out[M,128] = Powers[M,16] x W[16,128], W[k= t*4+p][o] = poly_coeffs[t][p][o]
// One wave32 computes a 16-token x 128-output tile with V_WMMA_F32_16X16X4_F32.
__global__ __launch_bounds__(256) void turn_embed_wmma_f32(
    const int* __restrict__ tok,      // (B*S) token ids
    const float* __restrict__ turns,  // (VOCAB, 4) fp32, 16B-aligned rows
    const float* __restrict__ pc,     // (4,4,128) fp32 == W[16][128] row-major
    float* __restrict__ out,          // (B*S, 128) fp32
    int num_tiles)
{
    const int lane = threadIdx.x & 31;
    const int lm   = lane & 15;      // matrix row (A) / col-in-tile (B,D)
    const int half = lane >> 4;      // 0: K local {0,1} / D rows 0-7 ; 1: K {2,3} / rows 8-15

    // ---- Preload B fragments (uniform across all tiles; 8KB table, L1/L2 hot)
    // B 4x16 fp32 layout: VGPR0 = K row (2*half+0), VGPR1 = K row (2*half+1), N = lm.
    v2f bfrag[NT][KSTEPS];
#pragma unroll
    for (int kk = 0; kk < KSTEPS; ++kk) {
        const int krow = kk * 4 + 2 * half;           // global k = t*4 + p
#pragma unroll
        for (int n = 0; n < NT; ++n) {
            const int col = n * 16 + lm;
            bfrag[n][kk][0] = pc[(krow + 0) * OUT_DIM + col];
            bfrag[n][kk][1] = pc[(krow + 1) * OUT_DIM + col];
        }
    }

    const int wave   = (int)((blockIdx.x * blockDim.x + threadIdx.x) >> 5);
    const int nwaves = (int)((gridDim.x * blockDim.x) >> 5);
    const v8f vzero = {};

    for (int tile = wave; tile < num_tiles; tile += nwaves) {
        const int row0 = tile * 16;

        // Gather this lane's token row of the turns table (both half-groups
        // read the same 16 rows; 16B global_load_b128 per lane, L2-resident).
        const int token = tok[row0 + lm];
        const v4f x4 = *(const v4f*)(turns + (size_t)token * 4);

        // Prefetch next tile's token ids (gfx1250 global_prefetch_b8).
        __builtin_prefetch(tok + row0 + nwaves * 16 + lm, 0, 3);

        v8f acc[NT];
#pragma unroll
        for (int n = 0; n < NT; ++n) acc[n] = vzero;

#pragma unroll
        for (int kk = 0; kk < KSTEPS; ++kk) {
            // A 16x4 fp32 fragment, exponent p = 2*half + j. Branchless: EXEC
            // stays all-ones around every WMMA (required by ISA).
            const float x  = x4[kk];
            const float x2 = x * x;
            v2f a;
            a[0] = half ? x2      : 1.0f;   // x^0 or x^2
            a[1] = half ? x2 * x  : x;      // x^1 or x^3
#pragma unroll
            for (int n = 0; n < NT; ++n) {
                // (neg_a, A, neg_b, B, c_mod, C, reuse_a, reuse_b)
                acc[n] = __builtin_amdgcn_wmma_f32_16x16x4_f32(
                    false, a, false, bfrag[n][kk], (short)0, acc[n],
                    false, false);
            }
        }

        // D 16x16 fp32 layout: VGPR r -> row (r + 8*half), col n*16 + lm.
        // 16-lane groups emit 64B-contiguous b32 stores; n-tiles fill lines.
        float* orow = out + (size_t)row0 * OUT_DIM;
#pragma unroll
        for (int n = 0; n < NT; ++n) {
            const int col = n * 16 + lm;
#pragma unroll
            for (int r = 0; r < 8; ++r) {
                orow[(size_t)(r + 8 * half) * OUT_DIM + col] = acc[n][r];
            }
        }
    }
}

extern "C" void kernel_launch(void* const* d_in, const int* in_sizes, int n_in,
                              void* d_out, int out_size, void* d_ws, size_t ws_size,
                              hipStream_t stream) {
    const int*   tok   = (const int*)d_in[0];    // (32, 8192) int32
    const float* turns = (const float*)d_in[1];  // (50257, 4) fp32
    const float* pc    = (const float*)d_in[2];  // (4, 4, 128) fp32
    float*       out   = (float*)d_out;          // (32, 8192, 128) fp32

    const int num_tokens = in_sizes[0];          // 262144
    const int num_tiles  = num_tokens / 16;      // 16384 (exact)

    // 512 blocks x 8 waves = 4096 waves; each wave handles 4 tiles via
    // grid-stride, amortizing the B-fragment preload.
    turn_embed_wmma_f32<<<512, 256, 0, stream>>>(tok, turns, pc, out, num_tiles);
}